// CudaDilation2D_40106404610410
// MI455X (gfx1250) — compile-verified
//
#include <hip/hip_runtime.h>
#include <stdint.h>

// ---------------------------------------------------------------------------
// Max-plus (tropical) dilation, CDNA5 / gfx1250, wave32.
//   out[b,o,y,x] = max_{c,i,j} f[b,c,y+i-2,x+j-2] + h[o,c,i,j]
// B=8, Cin=Cout=32, H=W=96, K=5.  VALU-bound (~1.9G add/max pairs vs ~20MB).
// WMMA is inapplicable (max-accumulate is not expressible as A*B+C), so we
// lean on gfx1250 async global->LDS DMA, scalar-broadcast h, and max3 fusion.
// ---------------------------------------------------------------------------

#define USE_ASYNC_LDS 1

namespace {
constexpr int Bn   = 8;
constexpr int Cin  = 32;
constexpr int Cout = 32;
constexpr int Hn   = 96;
constexpr int Wn   = 96;
constexpr int Kn   = 5;

constexpr int TILE   = 16;        // spatial tile (y and x)
constexpr int SZY    = TILE + 4;  // 20 rows incl. halo
constexpr int SZX    = 24;        // 16 + 2*2 halo, padded to 24 for 16B-aligned chunks
constexpr int ROWF   = SZX;       // floats per LDS row
constexpr int PLANEF = SZX * SZY;     // 480 floats per channel plane
constexpr int LDSF   = Cin * PLANEF;  // 15360 floats = 61440 B
constexpr int CHUNKS = LDSF / 4;      // 3840 float4 chunks
constexpr float NEGV = -1.0e38f;      // padding; always loses (center tap is real)
}

__global__ __launch_bounds__(256)
void dilate_maxplus_kernel(const float* __restrict__ f,
                           const float* __restrict__ h,
                           float* __restrict__ out)
{
    __shared__ __align__(16) float sf[LDSF];

    const int tid  = threadIdx.x;
    const int b    = blockIdx.y;
    const int tidx = blockIdx.x;
    const int tx0  = (tidx % (Wn / TILE)) * TILE;
    const int ty0  = (tidx / (Wn / TILE)) * TILE;

    // ---- Stage f[b, :, ty0-2 : ty0+18, tx0-4 : tx0+20] into LDS -------------
    // Interior 16B chunks go through the gfx1250 async LDS-DMA path; boundary
    // chunks are padded with NEGV via a predicated ds_store_b128 path.
    for (int k = tid; k < CHUNKS; k += 256) {
        const int c   = k / (PLANEF / 4);
        const int rem = k - c * (PLANEF / 4);
        const int r   = rem / (ROWF / 4);
        const int q   = rem - r * (ROWF / 4);
        const int gy  = ty0 - 2 + r;
        const int gx  = tx0 - 4 + q * 4;          // multiple of 4 -> 16B aligned
        const float* gsrc = f + (size_t)(b * Cin + c) * (Hn * Wn) + (gy * Wn + gx);
        float* ldst = &sf[k * 4];
        const bool full = (gy >= 0) && (gy < Hn) && (gx >= 0) && (gx + 4 <= Wn);
        if (full) {
#if USE_ASYNC_LDS
            unsigned lo = (unsigned)(uintptr_t)ldst;                 // LDS byte addr
            unsigned long long ga = (unsigned long long)(uintptr_t)gsrc;
            asm volatile("global_load_async_to_lds_b128 %0, %1, off"
                         :: "v"(lo), "v"(ga) : "memory");
#else
            *(float4*)ldst = *(const float4*)gsrc;
#endif
        } else {
            const bool yok = (gy >= 0) && (gy < Hn);
            float4 v;
            v.x = (yok && (gx + 0) >= 0 && (gx + 0) < Wn) ? gsrc[0] : NEGV;
            v.y = (yok && (gx + 1) >= 0 && (gx + 1) < Wn) ? gsrc[1] : NEGV;
            v.z = (yok && (gx + 2) >= 0 && (gx + 2) < Wn) ? gsrc[2] : NEGV;
            v.w = (yok && (gx + 3) >= 0 && (gx + 3) < Wn) ? gsrc[3] : NEGV;
            *(float4*)ldst = v;
        }
    }
#if USE_ASYNC_LDS
    asm volatile("s_wait_asynccnt 0" ::: "memory");
#endif
    __syncthreads();

    // ---- Compute: one output channel per wave (wave32), 8 px per lane ------
    const int lane = tid & 31;
    const int wid  = tid >> 5;                       // 0..7
    const int o    = __builtin_amdgcn_readfirstlane(blockIdx.z * 8 + wid);
    const int lx   = lane & 3;                       // col group (4 cols each)
    const int ly   = lane >> 2;                      // 0..7
    const int ry   = 2 * ly;                         // first owned tile row

    float acc0[4], acc1[4];
#pragma unroll
    for (int p = 0; p < 4; ++p) { acc0[p] = NEGV; acc1[p] = NEGV; }

    for (int c = 0; c < Cin; ++c) {
        // h taps for (o, c): wave-uniform -> scalar loads through K$.
        const float* __restrict__ hp = h + (size_t)(o * Cin + c) * (Kn * Kn);
        float hv[25];
#pragma unroll
        for (int t = 0; t < 25; ++t) hv[t] = hp[t];

        const float4* plane = (const float4*)&sf[c * PLANEF];
#pragma unroll
        for (int i = 0; i < Kn; ++i) {
            const int rb = (ry + i) * (ROWF / 4) + lx;     // ROWF/4 == 6
            const float4 a0 = plane[rb + 0];
            const float4 a1 = plane[rb + 1];
            const float4 a2 = plane[rb + 2];
            const float4 b0 = plane[rb + 6];
            const float4 b1 = plane[rb + 7];
            const float4 b2 = plane[rb + 8];
            const float av[12] = { a0.x, a0.y, a0.z, a0.w,
                                   a1.x, a1.y, a1.z, a1.w,
                                   a2.x, a2.y, a2.z, a2.w };
            const float bv[12] = { b0.x, b0.y, b0.z, b0.w,
                                   b1.x, b1.y, b1.z, b1.w,
                                   b2.x, b2.y, b2.z, b2.w };
#pragma unroll
            for (int p = 0; p < 4; ++p) {
                const int cb = 2 + p;                      // col offset in av/bv
                float s0 = acc0[p];
                s0 = fmaxf(fmaxf(s0, av[cb + 0] + hv[i * 5 + 0]),
                                     av[cb + 1] + hv[i * 5 + 1]);   // -> v_max3
                s0 = fmaxf(fmaxf(s0, av[cb + 2] + hv[i * 5 + 2]),
                                     av[cb + 3] + hv[i * 5 + 3]);   // -> v_max3
                s0 = fmaxf(s0, av[cb + 4] + hv[i * 5 + 4]);
                acc0[p] = s0;

                float s1 = acc1[p];
                s1 = fmaxf(fmaxf(s1, bv[cb + 0] + hv[i * 5 + 0]),
                                     bv[cb + 1] + hv[i * 5 + 1]);
                s1 = fmaxf(fmaxf(s1, bv[cb + 2] + hv[i * 5 + 2]),
                                     bv[cb + 3] + hv[i * 5 + 3]);
                s1 = fmaxf(s1, bv[cb + 4] + hv[i * 5 + 4]);
                acc1[p] = s1;
            }
        }
    }

    // ---- Store: 2 rows x 4 cols per lane, 16B-aligned b128 stores ----------
    const int ox = tx0 + 4 * lx;
    const int oy = ty0 + ry;
    float* op = out + ((size_t)(b * Cout + o) * Hn + oy) * Wn + ox;
    *(float4*)op        = make_float4(acc0[0], acc0[1], acc0[2], acc0[3]);
    *(float4*)(op + Wn) = make_float4(acc1[0], acc1[1], acc1[2], acc1[3]);
}

extern "C" void kernel_launch(void* const* d_in, const int* in_sizes, int n_in,
                              void* d_out, int out_size, void* d_ws, size_t ws_size,
                              hipStream_t stream)
{
    (void)in_sizes; (void)n_in; (void)out_size; (void)d_ws; (void)ws_size;
    const float* f = (const float*)d_in[0];
    const float* h = (const float*)d_in[1];
    float* out = (float*)d_out;

    dim3 grid((Hn / TILE) * (Wn / TILE),  // 36 spatial tiles
              Bn,                         // 8 batches
              Cout / 8);                  // 4 groups of 8 output channels
    dim3 block(256);
    dilate_maxplus_kernel<<<grid, block, 0, stream>>>(f, h, out);
}